// GEConv_32701880992136
// MI455X (gfx1250) — compile-verified
//
#include <hip/hip_runtime.h>
#include <math.h>

#define B_    16
#define C_    64
#define N_    2048
#define K_    20
#define S_    1024
#define OUT_  128
#define GE_   140            // 12 + 2*C
#define SK_   (S_*K_)        // 20480  (multiple of 16 -> col tiles never cross batch)
#define TOT_  (B_*SK_)       // 327680 columns of the conv GEMMs
#define NB_STAT 64

typedef float v2f __attribute__((ext_vector_type(2)));
typedef float v8f __attribute__((ext_vector_type(8)));

__device__ __forceinline__ v8f wmma_f32_4(v2f a, v2f b, v8f c) {
  // D = A(16x4 f32) * B(4x16 f32) + C(16x16 f32)
  return __builtin_amdgcn_wmma_f32_16x16x4_f32(false, a, false, b, (short)0, c, false, false);
}

// ---------------------------------------------------------------- squared norms
__global__ void xx_kernel(const float* __restrict__ x, float* __restrict__ xx) {
  int id = blockIdx.x * blockDim.x + threadIdx.x;
  if (id >= B_ * N_) return;
  int b = id / N_, n = id % N_;
  const float* xb = x + (size_t)b * C_ * N_ + n;
  float s = 0.f;
  #pragma unroll 8
  for (int c = 0; c < C_; ++c) { float v = xb[(size_t)c * N_]; s += v * v; }
  xx[id] = s;
}

// ---------------------------------------------------------------- farthest point sampling
__global__ void fps_kernel(const float* __restrict__ xyz, int* __restrict__ fps_idx,
                           float* __restrict__ nxyz, float* __restrict__ out_tail) {
  const int b = blockIdx.x;
  const int tid = threadIdx.x;
  const float* P = xyz + (size_t)b * 3 * N_;     // P[d*N + n]
  __shared__ float dist[N_];
  __shared__ float rv[256];
  __shared__ int   ri[256];
  __shared__ float cen[3];

  float s0 = 0.f, s1 = 0.f, s2 = 0.f;
  for (int n = tid; n < N_; n += 256) {
    s0 += P[n]; s1 += P[N_ + n]; s2 += P[2 * N_ + n];
    dist[n] = 1e10f;
  }
  rv[tid] = s0; __syncthreads();
  for (int o = 128; o; o >>= 1) { if (tid < o) rv[tid] += rv[tid + o]; __syncthreads(); }
  if (tid == 0) cen[0] = rv[0] / (float)N_;
  __syncthreads();
  rv[tid] = s1; __syncthreads();
  for (int o = 128; o; o >>= 1) { if (tid < o) rv[tid] += rv[tid + o]; __syncthreads(); }
  if (tid == 0) cen[1] = rv[0] / (float)N_;
  __syncthreads();
  rv[tid] = s2; __syncthreads();
  for (int o = 128; o; o >>= 1) { if (tid < o) rv[tid] += rv[tid + o]; __syncthreads(); }
  if (tid == 0) cen[2] = rv[0] / (float)N_;
  __syncthreads();

  for (int i = 0; i < S_; ++i) {
    float cx = cen[0], cy = cen[1], cz = cen[2];
    float bv = -1.f; int bi = 0;
    for (int n = tid; n < N_; n += 256) {
      float dx = P[n] - cx, dy = P[N_ + n] - cy, dz = P[2 * N_ + n] - cz;
      float d  = dx * dx + dy * dy + dz * dz;
      float dn = fminf(dist[n], d);
      dist[n] = dn;
      if (dn > bv) { bv = dn; bi = n; }     // strict > keeps lowest index (argmax tie rule)
    }
    rv[tid] = bv; ri[tid] = bi; __syncthreads();
    for (int o = 128; o; o >>= 1) {
      if (tid < o) {
        float v2 = rv[tid + o]; int i2 = ri[tid + o];
        if (v2 > rv[tid] || (v2 == rv[tid] && i2 < ri[tid])) { rv[tid] = v2; ri[tid] = i2; }
      }
      __syncthreads();
    }
    int far = ri[0];
    if (tid == 0) {
      fps_idx[b * S_ + i] = far;
      float px = P[far], py = P[N_ + far], pz = P[2 * N_ + far];
      float* nx = nxyz + ((size_t)b * S_ + i) * 3;
      nx[0] = px; nx[1] = py; nx[2] = pz;
      out_tail[(size_t)b * 3 * S_ + 0 * S_ + i] = px;
      out_tail[(size_t)b * 3 * S_ + 1 * S_ + i] = py;
      out_tail[(size_t)b * 3 * S_ + 2 * S_ + i] = pz;
      cen[0] = px; cen[1] = py; cen[2] = pz;
    }
    __syncthreads();
  }
}

// ---------------------------------------------------------------- KNN (rows at fps only) with WMMA
__global__ void knn_kernel(const float* __restrict__ x, const float* __restrict__ xx,
                           const int* __restrict__ fps_idx, int* __restrict__ idx_s) {
  const int bb   = blockIdx.x / (S_ / 16);
  const int st   = (blockIdx.x % (S_ / 16)) * 16;
  const int lane = threadIdx.x;
  const float* xb  = x  + (size_t)bb * C_ * N_;
  const float* xxb = xx + (size_t)bb * N_;

  __shared__ int   rowIdx[16];
  __shared__ float rowXX[16];
  __shared__ float sA[16][C_];        // gathered A rows, staged once (4 KB)
  __shared__ float sc[16][16];
  __shared__ float tv[16][K_];
  __shared__ int   ti[16][K_];

  if (lane < 16) {
    int r = fps_idx[bb * S_ + st + lane];
    rowIdx[lane] = r;
    rowXX[lane]  = xxb[r];
  }
  __syncthreads();
  // stage A rows: sA[row][c] = x[b][c][rowIdx[row]]
  for (int e = lane; e < 16 * C_; e += 32) {
    int row = e & 15, c = e >> 4;
    sA[row][c] = xb[(size_t)c * N_ + rowIdx[row]];
  }
  __syncthreads();

  const int l16  = lane & 15;
  const int half = lane >> 4;          // 0: lanes 0-15, 1: lanes 16-31
  int cnt = 0;

  for (int n0 = 0; n0 < N_; n0 += 16) {
    v8f acc = {0.f, 0.f, 0.f, 0.f, 0.f, 0.f, 0.f, 0.f};
    for (int kk = 0; kk < C_; kk += 4) {
      // A: lane-half picks K-pair {kk,kk+1} / {kk+2,kk+3} -> single b64 LDS load
      v2f a = *(const v2f*)&sA[l16][kk + 2 * half];
      // B: lane-half picks rows {kk,kk+2} / {kk+1,kk+3} -> base row kk+half, stride 2N
      const float* bp = xb + (size_t)(kk + half) * N_ + n0 + l16;
      v2f bf; bf.x = bp[0]; bf.y = bp[2 * N_];
      acc = wmma_f32_4(a, bf, acc);
    }
    #pragma unroll
    for (int r = 0; r < 8; ++r) {      // D layout: VGPR r -> rows r / r+8
      int m = r + half * 8;
      sc[m][l16] = 2.f * acc[r] - rowXX[m] - xxb[n0 + l16];
    }
    __syncthreads();
    if (lane < 16) {
      #pragma unroll 1
      for (int j = 0; j < 16; ++j) {
        float v = sc[lane][j];
        int   ci = n0 + j;
        if (cnt < K_) {
          int p = cnt;
          while (p > 0 && tv[lane][p - 1] < v) {
            tv[lane][p] = tv[lane][p - 1]; ti[lane][p] = ti[lane][p - 1]; --p;
          }
          tv[lane][p] = v; ti[lane][p] = ci; ++cnt;
        } else if (v > tv[lane][K_ - 1]) {
          int p = K_ - 1;
          while (p > 0 && tv[lane][p - 1] < v) {
            tv[lane][p] = tv[lane][p - 1]; ti[lane][p] = ti[lane][p - 1]; --p;
          }
          tv[lane][p] = v; ti[lane][p] = ci;
        }
      }
    }
    __syncthreads();
  }
  if (lane < 16) {
    int* dst = idx_s + ((size_t)(bb * S_ + st + lane)) * K_;
    for (int j = 0; j < K_; ++j) dst[j] = ti[lane][j];
  }
}

// ---------------------------------------------------------------- geometric features (pdx: [B,12,S,K])
__global__ void pdr_kernel(const float* __restrict__ xyz, const int* __restrict__ idx_s,
                           const float* __restrict__ nxyz, float* __restrict__ pdx) {
  const int b = blockIdx.x / S_;
  const int s = blockIdx.x % S_;
  const int lane = threadIdx.x;
  __shared__ float xk[K_][3];
  __shared__ float nxs[3];
  __shared__ float meanx[3], meanx2[3], nn[3];
  __shared__ float vals[6][K_];
  __shared__ float chm[6];

  if (lane < K_) {
    int idx = idx_s[((size_t)(b * S_ + s)) * K_ + lane];
    const float* P = xyz + (size_t)b * 3 * N_;
    xk[lane][0] = P[idx]; xk[lane][1] = P[N_ + idx]; xk[lane][2] = P[2 * N_ + idx];
  }
  if (lane < 3) nxs[lane] = nxyz[((size_t)(b * S_ + s)) * 3 + lane];
  __syncthreads();
  if (lane == 0) {
    float a0=0,a1=0,a2=0, c0=0,c1=0,c2=0;
    for (int k = 0; k < K_; ++k) {
      a0 += xk[k][0]-nxs[0]; a1 += xk[k][1]-nxs[1]; a2 += xk[k][2]-nxs[2];
      c0 += xk[k][0];        c1 += xk[k][1];        c2 += xk[k][2];
    }
    meanx[0]=a0/(float)K_; meanx[1]=a1/(float)K_; meanx[2]=a2/(float)K_;
    meanx2[0]=c0/(float)K_; meanx2[1]=c1/(float)K_; meanx2[2]=c2/(float)K_;
    nn[0]=nxs[0]-meanx2[0]; nn[1]=nxs[1]-meanx2[1]; nn[2]=nxs[2]-meanx2[2];
  }
  __syncthreads();
  if (lane < K_) {
    float nx0 = xk[lane][0]-nxs[0], nx1 = xk[lane][1]-nxs[1], nx2 = xk[lane][2]-nxs[2];
    float no0 = xk[lane][0]-meanx2[0], no1 = xk[lane][1]-meanx2[1], no2 = xk[lane][2]-meanx2[2];
    float p1 = sqrtf(nx0*nx0 + nx1*nx1 + nx2*nx2);
    float d0 = nx0-meanx[0], d1 = nx1-meanx[1], d2 = nx2-meanx[2];
    float p2 = sqrtf(d0*d0 + d1*d1 + d2*d2);
    float t0 = nx0+no0-nn[0], t1 = nx1+no1-nn[1], t2 = nx2+no2-nn[2];
    float p3 = sqrtf(t0*t0 + t1*t1 + t2*t2);
    float nmx = sqrtf(meanx[0]*meanx[0] + meanx[1]*meanx[1] + meanx[2]*meanx[2]);
    float nno = sqrtf(no0*no0 + no1*no1 + no2*no2);
    float nnn = sqrtf(nn[0]*nn[0] + nn[1]*nn[1] + nn[2]*nn[2]);
    float c1v = (nx0*meanx[0]+nx1*meanx[1]+nx2*meanx[2]) / (fmaxf(p1,1e-8f)*fmaxf(nmx,1e-8f));
    float c2v = (nx0*no0+nx1*no1+nx2*no2)                 / (fmaxf(p1,1e-8f)*fmaxf(nno,1e-8f));
    float c3v = (no0*nn[0]+no1*nn[1]+no2*nn[2])           / (fmaxf(nno,1e-8f)*fmaxf(nnn,1e-8f));
    vals[0][lane]=p1; vals[1][lane]=p2; vals[2][lane]=p3;
    vals[3][lane]=c1v; vals[4][lane]=c2v; vals[5][lane]=c3v;
  }
  __syncthreads();
  if (lane < 6) {
    float sum = 0.f;
    for (int k = 0; k < K_; ++k) sum += vals[lane][k];
    chm[lane] = sum / (float)K_;
  }
  __syncthreads();
  if (lane < K_) {
    size_t base = (((size_t)b * 12) * S_ + s) * K_ + lane;
    for (int ch = 0; ch < 6; ++ch) {
      pdx[base + (size_t)ch * SK_]       = vals[ch][lane];
      pdx[base + (size_t)(ch + 6) * SK_] = vals[ch][lane] - chm[ch];
    }
  }
}

// ---------------------------------------------------------------- conv1: y1 = W1(128x140) * g(140 x TOT), g built on the fly
__global__ void gemm1_kernel(const float* __restrict__ x, const float* __restrict__ W1,
                             const int* __restrict__ fps_idx, const int* __restrict__ idx_s,
                             const float* __restrict__ pdx, float* __restrict__ y1) {
  __shared__ float g[GE_][16];
  const int m0  = blockIdx.x * 16;
  const int b   = m0 / SK_;
  const int mmB = m0 % SK_;
  const float* xb  = x + (size_t)b * C_ * N_;
  const int*   fpb = fps_idx + b * S_;

  for (int e = threadIdx.x; e < GE_ * 16; e += 256) {
    int ch = e >> 4, col = e & 15;
    int mm = mmB + col;
    int s = mm / K_, k = mm % K_;
    float v;
    if (ch < 12) {
      v = pdx[(((size_t)b * 12 + ch) * S_ + s) * K_ + k];
    } else if (ch < 12 + C_) {
      int c = ch - 12;
      int lin = s * C_ + c;                      // faithful [B,C,S] -> [B,S,1,C] reinterpret
      float fq = xb[(size_t)(lin >> 10) * N_ + fpb[lin & 1023]];
      v = xb[(size_t)c * N_ + idx_s[((size_t)(b * S_ + s)) * K_ + k]] - fq;
    } else {
      int c = ch - 12 - C_;
      int lin = s * C_ + c;
      v = xb[(size_t)(lin >> 10) * N_ + fpb[lin & 1023]];
    }
    g[ch][col] = v;
  }
  __syncthreads();

  const int wave = threadIdx.x >> 5;
  const int lane = threadIdx.x & 31;
  const int l16  = lane & 15;
  const int half = lane >> 4;
  const int row0 = wave * 16;
  const int arow = row0 + l16;
  v8f acc = {0.f, 0.f, 0.f, 0.f, 0.f, 0.f, 0.f, 0.f};
  for (int kk = 0; kk < GE_; kk += 4) {
    v2f a = *(const v2f*)(W1 + arow * GE_ + kk + 2 * half);   // b64, branchless
    v2f bf; bf.x = g[kk + half][l16]; bf.y = g[kk + half + 2][l16];
    acc = wmma_f32_4(a, bf, acc);
  }
  const int rAdd = half * 8;
  #pragma unroll
  for (int r = 0; r < 8; ++r) {
    int row = row0 + r + rAdd;
    y1[((size_t)b * OUT_ + row) * SK_ + mmB + l16] = acc[r];
  }
}

// ---------------------------------------------------------------- conv2: out = W2(128x128) * h + b2
__global__ void gemm2_kernel(const float* __restrict__ h, const float* __restrict__ W2,
                             const float* __restrict__ b2, float* __restrict__ out) {
  __shared__ float g[OUT_][16];
  const int m0  = blockIdx.x * 16;
  const int b   = m0 / SK_;
  const int mmB = m0 % SK_;
  // stage B-panel with 128-bit loads/stores: 128 rows x 4 float4
  for (int e = threadIdx.x; e < OUT_ * 4; e += 256) {
    int ch = e >> 2, q = e & 3;
    const float4* src = (const float4*)(h + ((size_t)b * OUT_ + ch) * SK_ + mmB);
    *(float4*)&g[ch][q * 4] = src[q];
  }
  __syncthreads();
  const int wave = threadIdx.x >> 5;
  const int lane = threadIdx.x & 31;
  const int l16  = lane & 15;
  const int half = lane >> 4;
  const int row0 = wave * 16;
  const int arow = row0 + l16;
  v8f acc = {0.f, 0.f, 0.f, 0.f, 0.f, 0.f, 0.f, 0.f};
  for (int kk = 0; kk < OUT_; kk += 4) {
    v2f a = *(const v2f*)(W2 + arow * OUT_ + kk + 2 * half);  // b64, branchless
    v2f bf; bf.x = g[kk + half][l16]; bf.y = g[kk + half + 2][l16];
    acc = wmma_f32_4(a, bf, acc);
  }
  const int rAdd = half * 8;
  #pragma unroll
  for (int r = 0; r < 8; ++r) {
    int row = row0 + r + rAdd;
    out[((size_t)b * OUT_ + row) * SK_ + mmB + l16] = acc[r] + b2[row];
  }
}

// ---------------------------------------------------------------- BN stats (deterministic 2-stage, float4 streaming)
__global__ void bnstat1_kernel(const float* __restrict__ y, float* __restrict__ part) {
  const int ch = blockIdx.x / NB_STAT;
  const int j  = blockIdx.x % NB_STAT;
  const int tid = threadIdx.x;
  const int chunk  = TOT_ / NB_STAT;          // 5120 elems, contiguous within one (b,ch)
  const int chunk4 = chunk / 4;               // 1280 float4
  const int e0 = j * chunk;
  const int bb = e0 / SK_;                    // constant over chunk (5120 | 20480)
  const float4* base = (const float4*)(y + ((size_t)bb * OUT_ + ch) * SK_ + (e0 % SK_));
  float s = 0.f, s2 = 0.f;
  for (int e = tid; e < chunk4; e += 256) {
    __builtin_prefetch(base + e + 256, 0, 0);  // global_prefetch_b8 (speculative)
    float4 v = base[e];
    s  += v.x + v.y + v.z + v.w;
    s2 += v.x * v.x + v.y * v.y + v.z * v.z + v.w * v.w;
  }
  __shared__ float r1[256], r2[256];
  r1[tid] = s; r2[tid] = s2; __syncthreads();
  for (int o = 128; o; o >>= 1) {
    if (tid < o) { r1[tid] += r1[tid + o]; r2[tid] += r2[tid + o]; }
    __syncthreads();
  }
  if (tid == 0) {
    part[(ch * NB_STAT + j) * 2]     = r1[0];
    part[(ch * NB_STAT + j) * 2 + 1] = r2[0];
  }
}

__global__ void bnstat2_kernel(const float* __restrict__ part, float* __restrict__ stat) {
  const int ch = blockIdx.x;
  const int tid = threadIdx.x;  // 64 threads
  __shared__ float r1[64], r2[64];
  r1[tid] = part[(ch * NB_STAT + tid) * 2];
  r2[tid] = part[(ch * NB_STAT + tid) * 2 + 1];
  __syncthreads();
  for (int o = 32; o; o >>= 1) {
    if (tid < o) { r1[tid] += r1[tid + o]; r2[tid] += r2[tid + o]; }
    __syncthreads();
  }
  if (tid == 0) {
    float mu  = r1[0] / (float)TOT_;
    float var = r2[0] / (float)TOT_ - mu * mu;
    stat[ch * 2] = mu; stat[ch * 2 + 1] = var;
  }
}

// ---------------------------------------------------------------- BN apply + exact GELU (in place, float4)
__global__ void bngelu_kernel(float* __restrict__ y, const float* __restrict__ stat,
                              const float* __restrict__ gamma, const float* __restrict__ beta) {
  size_t id4 = (size_t)blockIdx.x * blockDim.x + threadIdx.x;
  if (id4 >= (size_t)B_ * OUT_ * SK_ / 4) return;
  int ch = (int)((id4 * 4 / SK_) % OUT_);     // SK_%4==0 -> whole float4 in one channel
  float4* p = (float4*)y + id4;
  float4 v = *p;
  float mu = stat[ch * 2], var = stat[ch * 2 + 1];
  float gm = gamma[ch], bt = beta[ch];
  float is = rsqrtf(var + 1e-5f);
  float t0 = gm * (v.x - mu) * is + bt;
  float t1 = gm * (v.y - mu) * is + bt;
  float t2 = gm * (v.z - mu) * is + bt;
  float t3 = gm * (v.w - mu) * is + bt;
  v.x = 0.5f * t0 * (1.f + erff(t0 * 0.7071067811865475f));
  v.y = 0.5f * t1 * (1.f + erff(t1 * 0.7071067811865475f));
  v.z = 0.5f * t2 * (1.f + erff(t2 * 0.7071067811865475f));
  v.w = 0.5f * t3 * (1.f + erff(t3 * 0.7071067811865475f));
  *p = v;
}

// ---------------------------------------------------------------- launch
extern "C" void kernel_launch(void* const* d_in, const int* in_sizes, int n_in,
                              void* d_out, int out_size, void* d_ws, size_t ws_size,
                              hipStream_t stream) {
  (void)in_sizes; (void)n_in; (void)out_size; (void)ws_size;
  const float* x   = (const float*)d_in[0];
  const float* xyz = (const float*)d_in[1];
  const float* W1  = (const float*)d_in[2];
  const float* g1  = (const float*)d_in[3];
  const float* be1 = (const float*)d_in[4];
  const float* W2  = (const float*)d_in[5];
  const float* b2  = (const float*)d_in[6];
  const float* g2  = (const float*)d_in[7];
  const float* be2 = (const float*)d_in[8];
  float* out = (float*)d_out;
  char*  ws  = (char*)d_ws;

  // workspace layout (bytes); total ~185.3 MB
  int*   fps  = (int*)  (ws + 0);          //  64 KB
  int*   idxs = (int*)  (ws + 65536);      //  1.25 MB
  float* xx   = (float*)(ws + 1376256);    //  128 KB
  float* nxyz = (float*)(ws + 1507328);    //  192 KB
  float* pdx  = (float*)(ws + 1703936);    //  15.7 MB
  float* y1   = (float*)(ws + 17432576);   //  167.8 MB  (h, reused in place)
  float* part = (float*)(ws + 185204736);  //  64 KB
  float* stat = (float*)(ws + 185270272);  //  1 KB

  float* out_tail = out + (size_t)B_ * OUT_ * SK_;   // nxyz^T [B,3,S]

  xx_kernel <<<(B_ * N_ + 255) / 256, 256, 0, stream>>>(x, xx);
  fps_kernel<<<B_, 256, 0, stream>>>(xyz, fps, nxyz, out_tail);
  knn_kernel<<<B_ * (S_ / 16), 32, 0, stream>>>(x, xx, fps, idxs);
  pdr_kernel<<<B_ * S_, 32, 0, stream>>>(xyz, idxs, nxyz, pdx);

  gemm1_kernel<<<TOT_ / 16, 256, 0, stream>>>(x, W1, fps, idxs, pdx, y1);
  bnstat1_kernel<<<OUT_ * NB_STAT, 256, 0, stream>>>(y1, part);
  bnstat2_kernel<<<OUT_, 64, 0, stream>>>(part, stat);
  bngelu_kernel<<<(int)(((size_t)B_ * OUT_ * SK_ / 4 + 255) / 256), 256, 0, stream>>>(y1, stat, g1, be1);

  gemm2_kernel<<<TOT_ / 16, 256, 0, stream>>>(y1, W2, b2, out);
  bnstat1_kernel<<<OUT_ * NB_STAT, 256, 0, stream>>>(out, part);
  bnstat2_kernel<<<OUT_, 64, 0, stream>>>(part, stat);
  bngelu_kernel<<<(int)(((size_t)B_ * OUT_ * SK_ / 4 + 255) / 256), 256, 0, stream>>>(out, stat, g2, be2);
}